// GINNet_6837587935809
// MI455X (gfx1250) — compile-verified
//
#include <hip/hip_runtime.h>
#include <hip/hip_bf16.h>

typedef __attribute__((ext_vector_type(2))) float v2f;
typedef __attribute__((ext_vector_type(8))) float v8f;

#define N_NODES 100000
#define N_EDGES 1200000
#define DIM 64
#define N_GRAPHS 512
#define N_OUT 10

// ---------------------------------------------------------------- zero fill
__global__ void k_zero(float* __restrict__ p, int n) {
  int t = blockIdx.x * blockDim.x + threadIdx.x;
  if (t < n) p[t] = 0.0f;
}

// ------------------------------------------------- edge scatter-add (agg[dst] += h[src])
__global__ void k_scatter(const float* __restrict__ h, const int* __restrict__ ei,
                          float* __restrict__ agg) {
  int t = blockIdx.x * blockDim.x + threadIdx.x;
  if (t >= N_EDGES * 16) return;
  int e = t >> 4;
  int c = (t & 15) << 2;
  int s = ei[e];
  int d = ei[N_EDGES + e];
  const float4 v = *(const float4*)(h + (size_t)s * DIM + c);
  float* p = agg + (size_t)d * DIM + c;
  atomicAdd(p + 0, v.x);
  atomicAdd(p + 1, v.y);
  atomicAdd(p + 2, v.z);
  atomicAdd(p + 3, v.w);
}

// ------------------------------------------------- GIN layer update:
//   z = (1+eps)*h + agg ; z = relu(bn1(z @ W1)) ; out = relu(bn2(z @ W2))
// One wave per 16 nodes, f32 WMMA 16x16x4, K=64 -> 16 steps per 16x16 tile.
__global__ __launch_bounds__(128) void k_gin(
    const float* __restrict__ h, const float* __restrict__ agg,
    float* __restrict__ out, const float* __restrict__ epsp,
    const float* __restrict__ W1,
    const float* __restrict__ g1, const float* __restrict__ be1,
    const float* __restrict__ m1, const float* __restrict__ v1,
    const float* __restrict__ W2,
    const float* __restrict__ g2, const float* __restrict__ be2,
    const float* __restrict__ m2, const float* __restrict__ v2) {
  __shared__ float sW1[64][65];
  __shared__ float sW2[64][65];
  __shared__ float sScale[2][64];
  __shared__ float sShift[2][64];
  __shared__ float sZ[4][16][68];

  const int tid = threadIdx.x;
  for (int i = tid; i < 4096; i += 128) {
    sW1[i >> 6][i & 63] = W1[i];
    sW2[i >> 6][i & 63] = W2[i];
  }
  if (tid < 64) {
    float sc = g1[tid] * rsqrtf(v1[tid] + 1e-5f);
    sScale[0][tid] = sc;
    sShift[0][tid] = be1[tid] - m1[tid] * sc;
  } else {
    int j = tid - 64;
    float sc = g2[j] * rsqrtf(v2[j] + 1e-5f);
    sScale[1][j] = sc;
    sShift[1][j] = be2[j] - m2[j] * sc;
  }
  const float ope = 1.0f + epsp[0];
  const int wave = tid >> 5;
  const int lane = tid & 31;
  const int tileBase = blockIdx.x * 64 + wave * 16;

  // stage z tile (16 nodes x 64 feats) into LDS, coalesced float4 loads
  for (int i = lane; i < 256; i += 32) {
    int r = i >> 4;
    int c = (i & 15) << 2;
    int node = tileBase + r;
    float4 hv = make_float4(0.f, 0.f, 0.f, 0.f);
    float4 av = make_float4(0.f, 0.f, 0.f, 0.f);
    if (node < N_NODES) {
      hv = *(const float4*)(h + (size_t)node * DIM + c);
      av = *(const float4*)(agg + (size_t)node * DIM + c);
    }
    sZ[wave][r][c + 0] = ope * hv.x + av.x;
    sZ[wave][r][c + 1] = ope * hv.y + av.y;
    sZ[wave][r][c + 2] = ope * hv.z + av.z;
    sZ[wave][r][c + 3] = ope * hv.w + av.w;
  }
  __syncthreads();

  // f32 WMMA lane mapping (ISA 7.12.2):
  //  A 16x4:  lanes 0-15 -> M=lane, K={4k,4k+1}; lanes 16-31 -> K={4k+2,4k+3}
  //  B 4x16:  lanes 0-15 -> N=lane, K={4k,4k+1}; lanes 16-31 -> N=lane-16, K={4k+2,4k+3}
  //  D 16x16: VGPR r -> M = r + 8*(lane>=16), N = lane&15
  const int row = lane & 15;
  const int koff = (lane >> 4) << 1;
  const int mbase = (lane >> 4) << 3;
  v8f acc[4];

  // ---------------- matmul 1: sZ @ W1 ----------------
#pragma unroll
  for (int ct = 0; ct < 4; ++ct) {
    int col = ct * 16 + row;
    v8f c = {0.f, 0.f, 0.f, 0.f, 0.f, 0.f, 0.f, 0.f};
#pragma unroll
    for (int kk = 0; kk < 16; ++kk) {
      int kb = (kk << 2) + koff;
      v2f a, b;
      a.x = sZ[wave][row][kb];
      a.y = sZ[wave][row][kb + 1];
      b.x = sW1[kb][col];
      b.y = sW1[kb + 1][col];
      c = __builtin_amdgcn_wmma_f32_16x16x4_f32(false, a, false, b, (short)0, c,
                                                false, false);
    }
    acc[ct] = c;
  }
  __syncthreads();
#pragma unroll
  for (int ct = 0; ct < 4; ++ct) {
    int n = ct * 16 + row;
    float sc = sScale[0][n], sh = sShift[0][n];
#pragma unroll
    for (int r = 0; r < 8; ++r) {
      sZ[wave][mbase + r][n] = fmaxf(acc[ct][r] * sc + sh, 0.0f);
    }
  }
  __syncthreads();

  // ---------------- matmul 2: sZ @ W2 ----------------
#pragma unroll
  for (int ct = 0; ct < 4; ++ct) {
    int col = ct * 16 + row;
    v8f c = {0.f, 0.f, 0.f, 0.f, 0.f, 0.f, 0.f, 0.f};
#pragma unroll
    for (int kk = 0; kk < 16; ++kk) {
      int kb = (kk << 2) + koff;
      v2f a, b;
      a.x = sZ[wave][row][kb];
      a.y = sZ[wave][row][kb + 1];
      b.x = sW2[kb][col];
      b.y = sW2[kb + 1][col];
      c = __builtin_amdgcn_wmma_f32_16x16x4_f32(false, a, false, b, (short)0, c,
                                                false, false);
    }
    acc[ct] = c;
  }
  __syncthreads();
#pragma unroll
  for (int ct = 0; ct < 4; ++ct) {
    int n = ct * 16 + row;
    float sc = sScale[1][n], sh = sShift[1][n];
#pragma unroll
    for (int r = 0; r < 8; ++r) {
      sZ[wave][mbase + r][n] = fmaxf(acc[ct][r] * sc + sh, 0.0f);
    }
  }
  __syncthreads();

  // coalesced float4 store
  for (int i = lane; i < 256; i += 32) {
    int r = i >> 4;
    int c = (i & 15) << 2;
    int node = tileBase + r;
    if (node < N_NODES) {
      float4 v = make_float4(sZ[wave][r][c + 0], sZ[wave][r][c + 1],
                             sZ[wave][r][c + 2], sZ[wave][r][c + 3]);
      *(float4*)(out + (size_t)node * DIM + c) = v;
    }
  }
}

// ------------------------------------------------- mean-pool scatter
__global__ void k_pool(const float* __restrict__ emb, const int* __restrict__ batch,
                       float* __restrict__ sums, float* __restrict__ cnt) {
  int t = blockIdx.x * blockDim.x + threadIdx.x;
  if (t >= N_NODES * 16) return;
  int node = t >> 4;
  int c = (t & 15) << 2;
  int g = batch[node];
  const float4 v = *(const float4*)(emb + (size_t)node * DIM + c);
  float* p = sums + (size_t)g * DIM + c;
  atomicAdd(p + 0, v.x);
  atomicAdd(p + 1, v.y);
  atomicAdd(p + 2, v.z);
  atomicAdd(p + 3, v.w);
  if ((t & 15) == 0) atomicAdd(cnt + g, 1.0f);
}

// ------------------------------------------------- MLP head + softmax (1 thread / graph)
__global__ __launch_bounds__(128) void k_head(
    const float* __restrict__ sums, const float* __restrict__ cnt,
    const float* __restrict__ W0, const float* __restrict__ b0,
    const float* __restrict__ W1, const float* __restrict__ b1,
    float* __restrict__ logits, float* __restrict__ probs) {
  int g = blockIdx.x * blockDim.x + threadIdx.x;
  if (g >= N_GRAPHS) return;
  float inv = 1.0f / fmaxf(cnt[g], 1.0f);
  float pooled[DIM];
#pragma unroll
  for (int j = 0; j < DIM; ++j) pooled[j] = sums[(size_t)g * DIM + j] * inv;
  float hdd[DIM];
#pragma unroll 4
  for (int k = 0; k < DIM; ++k) {
    float s = b0[k];
#pragma unroll
    for (int j = 0; j < DIM; ++j) s += pooled[j] * W0[j * DIM + k];
    hdd[k] = s > 0.0f ? s : (expf(s) - 1.0f);  // ELU
  }
  float lg[N_OUT];
  float mx = -1e30f;
#pragma unroll
  for (int o = 0; o < N_OUT; ++o) {
    float s = b1[o];
#pragma unroll 8
    for (int k = 0; k < DIM; ++k) s += hdd[k] * W1[k * N_OUT + o];
    lg[o] = s;
    mx = fmaxf(mx, s);
  }
  float se = 0.0f;
  float ex[N_OUT];
#pragma unroll
  for (int o = 0; o < N_OUT; ++o) {
    ex[o] = expf(lg[o] - mx);
    se += ex[o];
  }
  float rse = 1.0f / se;
#pragma unroll
  for (int o = 0; o < N_OUT; ++o) {
    logits[g * N_OUT + o] = lg[o];
    probs[g * N_OUT + o] = ex[o] * rse;
  }
}

// ================================================================ host
extern "C" void kernel_launch(void* const* d_in, const int* in_sizes, int n_in,
                              void* d_out, int out_size, void* d_ws, size_t ws_size,
                              hipStream_t stream) {
  // ---- locate inputs by unique size; collect param leaves in order ----
  const float* x = nullptr;
  const int* ei = nullptr;
  const int* batch = nullptr;
  int pidx[64];
  int np = 0;
  for (int i = 0; i < n_in; ++i) {
    int s = in_sizes[i];
    if (s == N_NODES * DIM) x = (const float*)d_in[i];
    else if (s == 2 * N_EDGES) ei = (const int*)d_in[i];
    else if (s == N_NODES) batch = (const int*)d_in[i];
    else if (np < 64) pidx[np++] = i;
  }

  struct Lyr { const float *eps, *W1, *W2, *g1, *be1, *m1, *v1, *g2, *be2, *m2, *v2; } L[3];
  int p = 0;
  auto nxt = [&]() -> const float* { return (const float*)d_in[pidx[p++]]; };
  // sorted pytree order starts each layer with W1 (4096); insertion order with eps (1)
  bool sortedOrder = (np > 0 && in_sizes[pidx[0]] == DIM * DIM);
  for (int l = 0; l < 3; ++l) {
    if (sortedOrder) {  // W1, W2, bn1{beta,gamma,mean,var}, bn2{...}, eps
      L[l].W1 = nxt(); L[l].W2 = nxt();
      L[l].be1 = nxt(); L[l].g1 = nxt(); L[l].m1 = nxt(); L[l].v1 = nxt();
      L[l].be2 = nxt(); L[l].g2 = nxt(); L[l].m2 = nxt(); L[l].v2 = nxt();
      L[l].eps = nxt();
    } else {            // eps, W1, bn1{gamma,beta,mean,var}, W2, bn2{...}
      L[l].eps = nxt(); L[l].W1 = nxt();
      L[l].g1 = nxt(); L[l].be1 = nxt(); L[l].m1 = nxt(); L[l].v1 = nxt();
      L[l].W2 = nxt();
      L[l].g2 = nxt(); L[l].be2 = nxt(); L[l].m2 = nxt(); L[l].v2 = nxt();
    }
  }
  const float* mW0 = nxt();
  const float* mb0 = nxt();
  const float* mW1h = nxt();
  const float* mb1h = nxt();

  // ---- output & workspace layout ----
  float* logits = (float*)d_out;                         // [G, OUT]
  float* probs = logits + N_GRAPHS * N_OUT;              // [G, OUT]
  float* emb = probs + N_GRAPHS * N_OUT;                 // [N, D] (also h buffer)
  float* agg = (float*)d_ws;                             // [N, D]
  float* sums = agg + (size_t)N_NODES * DIM;             // [G, D]
  float* cntb = sums + (size_t)N_GRAPHS * DIM;           // [G]

  const int aggN = N_NODES * DIM;
  dim3 blk(256);
  dim3 zgrid((aggN + 255) / 256);
  dim3 sgrid((N_EDGES * 16 + 255) / 256);
  dim3 ggrid((N_NODES + 63) / 64), gblk(128);

  const float* hin = x;
  for (int l = 0; l < 3; ++l) {
    k_zero<<<zgrid, blk, 0, stream>>>(agg, aggN);
    k_scatter<<<sgrid, blk, 0, stream>>>(hin, ei, agg);
    k_gin<<<ggrid, gblk, 0, stream>>>(hin, agg, emb, L[l].eps,
                                      L[l].W1, L[l].g1, L[l].be1, L[l].m1, L[l].v1,
                                      L[l].W2, L[l].g2, L[l].be2, L[l].m2, L[l].v2);
    hin = emb;  // layers 2/3 run in place on emb (tile staged in LDS first)
  }

  int poolN = N_GRAPHS * DIM + N_GRAPHS;  // sums + cnt contiguous
  k_zero<<<dim3((poolN + 255) / 256), blk, 0, stream>>>(sums, poolN);
  k_pool<<<dim3((N_NODES * 16 + 255) / 256), blk, 0, stream>>>(emb, batch, sums, cntb);
  k_head<<<dim3((N_GRAPHS + 127) / 128), dim3(128), 0, stream>>>(sums, cntb, mW0, mb0,
                                                                 mW1h, mb1h, logits, probs);
}